// IMITEREmbeddings_19378892440163
// MI455X (gfx1250) — compile-verified
//
#include <hip/hip_runtime.h>
#include <hip/hip_bf16.h>
#include <cstddef>
#include <cstdint>

// ---------------------------------------------------------------------------
// IMITER embeddings for MI455X (gfx1250, wave32, WMMA + async-to-LDS)
//   kernel 1: patch-embedding conv as WMMA f32 GEMM (36864x768x768),
//             double-buffered LDS tiles filled with
//             global_load_async_to_lds_b128 (ASYNCcnt), epilogue fused with
//             conv bias + vis_pos_emb + mod_type_emb[1]
//   kernel 2: text embedding gather + LayerNorm + mod_type_emb[0]
//   kernel 3: cls row broadcast + mask vector
// ---------------------------------------------------------------------------

#define BATCH    64
#define LSEQ     40
#define DIM      768
#define NPATCH   576          // 24*24
#define SEQ      617          // 1 + 40 + 576
#define KTOT     768          // 3*16*16
#define EMB_ELEMS ((size_t)BATCH * SEQ * DIM)

typedef __attribute__((ext_vector_type(2))) float v2f;
typedef __attribute__((ext_vector_type(8))) float v8f;

#if defined(__HIP_DEVICE_COMPILE__)
#if !__has_builtin(__builtin_amdgcn_wmma_f32_16x16x4_f32)
#error "gfx1250 f32 WMMA builtin __builtin_amdgcn_wmma_f32_16x16x4_f32 not available on device"
#endif
#endif

// Async copy 16B global -> LDS, tracked by ASYNCcnt (CDNA5 VGLOBAL opcode 98).
__device__ __forceinline__ void async_copy_b128(uint32_t lds_addr, const float* gptr)
{
    asm volatile("global_load_async_to_lds_b128 %0, %1, off"
                 :: "v"(lds_addr), "v"((uint64_t)(uintptr_t)gptr)
                 : "memory");
}
__device__ __forceinline__ void wait_async_le8()
{
    asm volatile("s_wait_asynccnt 0x8" ::: "memory");
}
__device__ __forceinline__ void wait_async_0()
{
    asm volatile("s_wait_asynccnt 0x0" ::: "memory");
}

// ---------------------------------------------------------------------------
// Patch-embedding GEMM.
//   A[m,k]  : pixel_values, m = b*576 + ph*24 + pw, k = c*256 + i*16 + j
//   B[k,n]  : conv_w[n,k] (conv_w is (D, C*16*16) row-major)
//   grid (576, 12), block 128 threads = 4 waves; block tile 64x64;
//   wave w computes rows [w*16, w*16+16) x all 64 columns.
// ---------------------------------------------------------------------------

// Issue the 8 async b128 copies (4 A-slots + 4 B-slots) for one 64x32 K-tile.
__device__ __forceinline__ void stage_tile(int kBase, int tid, int M0, int N0,
                                           const float* __restrict__ px,
                                           const float* __restrict__ w,
                                           float (*A)[36], float (*B)[36])
{
#pragma unroll
    for (int s = 0; s < 4; ++s) {
        const int slot = tid + s * 128;   // 512 float4 slots per matrix
        const int row  = slot >> 3;       // 0..63
        const int c4   = slot & 7;        // 0..7
        const int k    = kBase + c4 * 4;

        // A: decode (m,k) -> NCHW pixel address (float4 along j, 16B aligned)
        const int m  = M0 + row;
        const int b  = m / NPATCH;
        const int p  = m - b * NPATCH;
        const int ph = p / 24;
        const int pw = p - ph * 24;
        const int c  = k >> 8;
        const int r8 = k & 255;
        const int i  = r8 >> 4;
        const int j  = r8 & 15;
        async_copy_b128((uint32_t)(uintptr_t)&A[row][c4 * 4],
                        px + (((size_t)(b * 3 + c) * 384 + ph * 16 + i) * 384 +
                              pw * 16 + j));

        // B: conv_w is contiguous in k for fixed n
        async_copy_b128((uint32_t)(uintptr_t)&B[row][c4 * 4],
                        w + (size_t)(N0 + row) * KTOT + k);
    }
}

__global__ __launch_bounds__(128)
void patch_embed_gemm(const float* __restrict__ px,    // (64,3,384,384)
                      const float* __restrict__ w,     // (768,768)  [n][k]
                      const float* __restrict__ bias,  // (768)
                      const float* __restrict__ vpos,  // (1,577,768)
                      const float* __restrict__ mod,   // (2,768)
                      float* __restrict__ out)         // emb (64,617,768)
{
    // stride 36 floats: rows 16B aligned (for b128 LDS writes); fragment-read
    // banks = 4*(9m mod 16) -> 16 distinct banks per half-wave, halves disjoint.
    __shared__ float Alds[2][64][36];
    __shared__ float Blds[2][64][36];

    const int tid  = threadIdx.x;
    const int lane = tid & 31;
    const int wave = tid >> 5;
    const int M0   = blockIdx.x * 64;
    const int N0   = blockIdx.y * 64;

    v8f acc[4];
#pragma unroll
    for (int i = 0; i < 4; ++i) acc[i] = (v8f){};

    const int mloc = wave * 16 + (lane & 15); // A-fragment row within block tile
    const int ksel = (lane >> 4) << 1;        // lanes 0-15 -> K+0/1, 16-31 -> K+2/3

    // Prologue: stage tile 0 into buffer 0.
    stage_tile(0, tid, M0, N0, px, w, Alds[0], Blds[0]);

    for (int t = 0; t < KTOT / 32; ++t) {
        const int cur = t & 1;
        if (t + 1 < KTOT / 32) {
            // Start tile t+1 into the other buffer (last read in the previous iter,
            // protected by the end-of-iteration barrier below).
            stage_tile((t + 1) * 32, tid, M0, N0, px, w,
                       Alds[cur ^ 1], Blds[cur ^ 1]);
            wait_async_le8();   // in-order ASYNCcnt: tile t's 8 copies done
        } else {
            wait_async_0();
        }
        __syncthreads();        // all waves' tile-t data visible in LDS

        const float (*Ab)[36] = Alds[cur];
        const float (*Bb)[36] = Blds[cur];

#if defined(__HIP_DEVICE_COMPILE__)
#pragma unroll
        for (int kb = 0; kb < 8; ++kb) {
            const int kk = kb * 4 + ksel;
            const float2 af = *(const float2*)&Ab[mloc][kk];
            v2f a; a.x = af.x; a.y = af.y;
#pragma unroll
            for (int nt = 0; nt < 4; ++nt) {
                const float2 bf = *(const float2*)&Bb[nt * 16 + (lane & 15)][kk];
                v2f bb; bb.x = bf.x; bb.y = bf.y;
                acc[nt] = __builtin_amdgcn_wmma_f32_16x16x4_f32(
                    /*neg_a=*/false, a, /*neg_b=*/false, bb,
                    /*c_mod=*/(short)0, acc[nt],
                    /*reuse_a=*/false, /*reuse_b=*/false);
            }
        }
#endif
        __syncthreads();        // done reading buf[cur] before iter t+1 refills it
    }

    // ---- epilogue: + conv_b + vis_pos_emb[p+1] + mod_type_emb[1], scatter --
    // C/D layout: VGPR r -> M = r (lanes 0-15) / r+8 (lanes 16-31), N = lane&15
    const int mbase = M0 + wave * 16 + ((lane >> 4) << 3);
#pragma unroll
    for (int nt = 0; nt < 4; ++nt) {
        const int n    = N0 + nt * 16 + (lane & 15);
        const float cn = bias[n] + mod[DIM + n];
#pragma unroll
        for (int r = 0; r < 8; ++r) {
            const int m = mbase + r;
            const int b = m / NPATCH;
            const int p = m - b * NPATCH;
            const float val = acc[nt][r] + cn + vpos[(size_t)(p + 1) * DIM + n];
            out[((size_t)b * SEQ + 1 + LSEQ + p) * DIM + n] = val;
        }
    }
}

// ---------------------------------------------------------------------------
// Text embeddings: gather + add + LayerNorm + mod_type_emb[0]
// One block (256 threads, 8 waves) per (b,l) token; 3 elements per thread.
// ---------------------------------------------------------------------------
__global__ __launch_bounds__(256)
void text_embed_ln(const int*   __restrict__ ids,
                   const int*   __restrict__ ttids,
                   const float* __restrict__ we,    // (V,768)
                   const float* __restrict__ pe,    // (40,768)
                   const float* __restrict__ tte,   // (2,768)
                   const float* __restrict__ g,
                   const float* __restrict__ be,
                   const float* __restrict__ mod,   // (2,768)
                   float* __restrict__ out)
{
    const int bl = blockIdx.x;            // 0..2559
    const int b  = bl / LSEQ;
    const int l  = bl - b * LSEQ;
    const float* wrow = we  + (size_t)ids[bl]   * DIM;
    const float* prow = pe  + (size_t)l         * DIM;
    const float* trow = tte + (size_t)ttids[bl] * DIM;

    float v[3];
    float s = 0.f, sq = 0.f;
#pragma unroll
    for (int e = 0; e < 3; ++e) {
        const int d = threadIdx.x + e * 256;
        const float x = wrow[d] + prow[d] + trow[d];
        v[e] = x;
        s += x;
        sq += x * x;
    }

    // wave32 butterfly, then cross-wave via LDS
    __shared__ float ssum[8], ssq[8];
#pragma unroll
    for (int off = 16; off; off >>= 1) {
        s  += __shfl_xor(s, off, 32);
        sq += __shfl_xor(sq, off, 32);
    }
    const int wave = threadIdx.x >> 5, lane = threadIdx.x & 31;
    if (lane == 0) { ssum[wave] = s; ssq[wave] = sq; }
    __syncthreads();
    if (wave == 0) {
        float a = (lane < 8) ? ssum[lane] : 0.f;
        float c = (lane < 8) ? ssq[lane]  : 0.f;
#pragma unroll
        for (int off = 4; off; off >>= 1) {
            a += __shfl_xor(a, off, 32);
            c += __shfl_xor(c, off, 32);
        }
        if (lane == 0) { ssum[0] = a; ssq[0] = c; }
    }
    __syncthreads();

    const float mean = ssum[0] * (1.f / DIM);
    const float var  = ssq[0] * (1.f / DIM) - mean * mean;
    const float inv  = rsqrtf(var + 1e-12f);

    float* orow = out + ((size_t)b * SEQ + 1 + l) * DIM;
#pragma unroll
    for (int e = 0; e < 3; ++e) {
        const int d = threadIdx.x + e * 256;
        orow[d] = (v[e] - mean) * inv * g[d] + be[d] + mod[d];
    }
}

// ---------------------------------------------------------------------------
// cls row broadcast + mask vector (masks appended after emb in d_out, as f32)
// ---------------------------------------------------------------------------
__global__ void cls_mask_kernel(const float* __restrict__ cls,
                                const int*   __restrict__ amask,
                                float* __restrict__ out)
{
    const int idx = blockIdx.x * blockDim.x + threadIdx.x;
    const int CLS_N  = BATCH * DIM;
    const int MASK_N = BATCH * SEQ;
    if (idx < CLS_N) {
        const int b = idx / DIM, d = idx - b * DIM;
        out[(size_t)b * SEQ * DIM + d] = cls[d];
    } else if (idx < CLS_N + MASK_N) {
        const int m = idx - CLS_N;
        const int b = m / SEQ, sp = m - b * SEQ;
        float mv = 1.0f;
        if (sp >= 1 && sp <= LSEQ) mv = (float)amask[b * LSEQ + sp - 1];
        out[EMB_ELEMS + m] = mv;
    }
}

// ---------------------------------------------------------------------------
extern "C" void kernel_launch(void* const* d_in, const int* in_sizes, int n_in,
                              void* d_out, int out_size, void* d_ws, size_t ws_size,
                              hipStream_t stream)
{
    (void)in_sizes; (void)n_in; (void)out_size; (void)d_ws; (void)ws_size;

    const int*   input_ids    = (const int*)  d_in[0];
    const int*   attn_mask    = (const int*)  d_in[1];
    const int*   token_type   = (const int*)  d_in[2];
    const float* pixel_values = (const float*)d_in[3];
    // d_in[4] = pixel_mask (unused by reference math)
    const float* word_emb     = (const float*)d_in[5];
    const float* text_pos_emb = (const float*)d_in[6];
    const float* text_tok_emb = (const float*)d_in[7];
    const float* ln_g         = (const float*)d_in[8];
    const float* ln_b         = (const float*)d_in[9];
    const float* conv_w       = (const float*)d_in[10];
    const float* conv_b       = (const float*)d_in[11];
    const float* cls_token    = (const float*)d_in[12];
    const float* vis_pos_emb  = (const float*)d_in[13];
    const float* mod_type_emb = (const float*)d_in[14];

    float* out = (float*)d_out;

    // 1) vision patch-embedding GEMM via WMMA (dominant work)
    dim3 gGemm(576, 12);                 // M tiles (36864/64) x N tiles (768/64)
    patch_embed_gemm<<<gGemm, 128, 0, stream>>>(
        pixel_values, conv_w, conv_b, vis_pos_emb, mod_type_emb, out);

    // 2) text embeddings + LayerNorm
    text_embed_ln<<<BATCH * LSEQ, 256, 0, stream>>>(
        input_ids, token_type, word_emb, text_pos_emb, text_tok_emb,
        ln_g, ln_b, mod_type_emb, out);

    // 3) cls row + masks
    const int total = BATCH * DIM + BATCH * SEQ;
    cls_mask_kernel<<<(total + 255) / 256, 256, 0, stream>>>(
        cls_token, attn_mask, out);
}